// GCN_57174604644381
// MI455X (gfx1250) — compile-verified
//
#include <hip/hip_runtime.h>
#include <cstddef>

typedef __attribute__((ext_vector_type(2))) float v2f;
typedef __attribute__((ext_vector_type(8))) float v8f;

#define GCN_N 50000

// ---------------------------------------------------------------------------
// GEMM: Y[N,OUTD] = (RELU ? max(X,0) : X)[N,128] @ W[128,OUTD]
// One block per 16-row strip. Wave w computes 16x16 tile at columns w*16.
// A strip staged in LDS (stride 132 floats -> conflict-free fragment reads).
// Uses V_WMMA_F32_16X16X4_F32, 32 WMMA per tile over K=128.
// ---------------------------------------------------------------------------
template <int OUTD, bool RELU>
__global__ __launch_bounds__(256) void gemm_wmma_k(const float* __restrict__ X,
                                                   const float* __restrict__ W,
                                                   float* __restrict__ Y) {
  __shared__ float sA[16 * 132];  // 16 rows x 128 cols, +4 pad
  const int tid  = threadIdx.x;
  const int lane = tid & 31;
  const int wave = tid >> 5;
  const int row0 = blockIdx.x * 16;

  // Coalesced stage of the 16x128 fp32 strip (512 float4), ReLU fused.
  const float4* Xv = (const float4*)(X + (size_t)row0 * 128);
  for (int i = tid; i < 512; i += blockDim.x) {
    const int r  = i >> 5;      // row in strip
    const int c4 = i & 31;      // float4 column
    float4 v = Xv[r * 32 + c4];
    if (RELU) {
      v.x = fmaxf(v.x, 0.0f); v.y = fmaxf(v.y, 0.0f);
      v.z = fmaxf(v.z, 0.0f); v.w = fmaxf(v.w, 0.0f);
    }
    *(float4*)(&sA[r * 132 + c4 * 4]) = v;
  }
  __syncthreads();

  // Fragment layout (ISA 7.12.2, 32-bit A 16x4 / B 4x16):
  //   lanes 0-15 hold K = k0+{0,1}, lanes 16-31 hold K = k0+{2,3}
  const int m   = lane & 15;            // row (A) / column (B,D) index
  const int kk  = (lane >> 4) << 1;     // 0 or 2
  const int col0 = wave * 16;
  const float* srow = &sA[m * 132];

  v8f acc = {};
#pragma unroll 4
  for (int k0 = 0; k0 < 128; k0 += 4) {
    v2f a, b;
    a.x = srow[k0 + kk];
    a.y = srow[k0 + kk + 1];
    b.x = W[(size_t)(k0 + kk) * OUTD + col0 + m];
    b.y = W[(size_t)(k0 + kk + 1) * OUTD + col0 + m];
    acc = __builtin_amdgcn_wmma_f32_16x16x4_f32(
        /*neg_a=*/false, a, /*neg_b=*/false, b,
        /*c_mod=*/(short)0, acc, /*reuse_a=*/false, /*reuse_b=*/false);
  }

  // D layout: VGPR r -> M = r (lanes 0-15), M = r+8 (lanes 16-31); N = lane&15
  const int orow = row0 + (lane >> 4) * 8;
#pragma unroll
  for (int r = 0; r < 8; ++r)
    Y[(size_t)(orow + r) * OUTD + col0 + m] = acc[r];
}

// ---------------------------------------------------------------------------
// agg[i*D + f] = bias[f]  (folds the "+ b" into the scatter target)
// ---------------------------------------------------------------------------
template <int D>
__global__ void init_bias_k(float* __restrict__ agg, const float* __restrict__ b,
                            int total) {
  const int i = blockIdx.x * blockDim.x + threadIdx.x;
  if (i < total) agg[i] = b[i & (D - 1)];
}

// ---------------------------------------------------------------------------
// Edge scatter: agg[dst[e]] += w[e] * sup[src[e]]
// D=128: 1 edge per wave (lane = feature quad). D=32: 4 edges per wave.
// Gather rows are L2-resident (25.6 MB << 192 MB L2).
// ---------------------------------------------------------------------------
template <int D>
__global__ __launch_bounds__(256) void scatter_edges_k(
    const float* __restrict__ sup, const int* __restrict__ src,
    const int* __restrict__ dst, const float* __restrict__ ew,
    float* __restrict__ agg, int nedges) {
  constexpr int F4  = D / 4;    // float4 chunks per row
  constexpr int EPW = 32 / F4;  // edges per wave
  const int lane   = threadIdx.x & 31;
  const int waveId = (blockIdx.x * blockDim.x + threadIdx.x) >> 5;
  const int sub = lane / F4;
  const int f4  = lane % F4;
  const long long e = (long long)waveId * EPW + sub;
  if (e >= nedges) return;

  const int   s = src[e];
  const int   d = dst[e];
  const float w = ew[e];
  const float4 v = *(const float4*)(sup + (size_t)s * D + f4 * 4);
  float* o = agg + (size_t)d * D + f4 * 4;
  atomicAdd(o + 0, w * v.x);
  atomicAdd(o + 1, w * v.y);
  atomicAdd(o + 2, w * v.z);
  atomicAdd(o + 3, w * v.w);
}

// ---------------------------------------------------------------------------
// In-place log_softmax over 32 classes: one wave32 per row, lane = class.
// ---------------------------------------------------------------------------
__global__ __launch_bounds__(256) void log_softmax32_k(float* __restrict__ h,
                                                       int nrows) {
  const int wid  = (blockIdx.x * blockDim.x + threadIdx.x) >> 5;
  const int lane = threadIdx.x & 31;
  if (wid >= nrows) return;
  const size_t idx = (size_t)wid * 32 + lane;
  const float v = h[idx];
  float m = v;
#pragma unroll
  for (int o = 16; o > 0; o >>= 1) m = fmaxf(m, __shfl_xor(m, o, 32));
  float s = expf(v - m);
#pragma unroll
  for (int o = 16; o > 0; o >>= 1) s += __shfl_xor(s, o, 32);
  h[idx] = v - m - logf(s);
}

// ---------------------------------------------------------------------------
extern "C" void kernel_launch(void* const* d_in, const int* in_sizes, int n_in,
                              void* d_out, int out_size, void* d_ws,
                              size_t ws_size, hipStream_t stream) {
  const float* x  = (const float*)d_in[0];
  const int*   ei = (const int*)d_in[1];
  const float* ew = (const float*)d_in[2];
  const float* W1 = (const float*)d_in[3];
  const float* b1 = (const float*)d_in[4];
  const float* W2 = (const float*)d_in[5];
  const float* b2 = (const float*)d_in[6];
  const float* W3 = (const float*)d_in[7];
  const float* b3 = (const float*)d_in[8];

  const int N = GCN_N;
  const int E = in_sizes[1] / 2;          // edge_index is [2, E]
  const int* src = ei;
  const int* dst = ei + E;
  float* out = (float*)d_out;

  float* bufA = (float*)d_ws;             // support   [N,128] (25.6 MB)
  float* bufB = bufA + (size_t)N * 128;   // agg / h   [N,128] (25.6 MB)

  const dim3 gGemm(N / 16);               // 3125 blocks, exact
  const int initBlocks128 = (N * 128 + 255) / 256;
  const int initBlocks32  = (N * 32 + 255) / 256;
  // scatter: 8 waves/block -> 8*EPW edges per block
  const int scatBlocks128 = (E + 8 * 1 - 1) / (8 * 1);   // EPW=1 for D=128
  const int scatBlocks32  = (E + 8 * 4 - 1) / (8 * 4);   // EPW=4 for D=32
  const int smBlocks      = (N + 7) / 8;                 // 8 rows per block

  // ---- Layer 1: h1 = relu(scatter(x@W1) + b1) (relu deferred to next load)
  gemm_wmma_k<128, false><<<gGemm, 256, 0, stream>>>(x, W1, bufA);
  init_bias_k<128><<<initBlocks128, 256, 0, stream>>>(bufB, b1, N * 128);
  scatter_edges_k<128><<<scatBlocks128, 256, 0, stream>>>(bufA, src, dst, ew,
                                                          bufB, E);

  // ---- Layer 2: h2 = relu(scatter(relu(h1)@W2) + b2)
  gemm_wmma_k<128, true><<<gGemm, 256, 0, stream>>>(bufB, W2, bufA);
  init_bias_k<128><<<initBlocks128, 256, 0, stream>>>(bufB, b2, N * 128);
  scatter_edges_k<128><<<scatBlocks128, 256, 0, stream>>>(bufA, src, dst, ew,
                                                          bufB, E);

  // ---- Layer 3: out = log_softmax(scatter(relu(h2)@W3) + b3)
  gemm_wmma_k<32, true><<<gGemm, 64, 0, stream>>>(bufB, W3, bufA);
  init_bias_k<32><<<initBlocks32, 256, 0, stream>>>(out, b3, N * 32);
  scatter_edges_k<32><<<scatBlocks32, 256, 0, stream>>>(bufA, src, dst, ew,
                                                        out, E);
  log_softmax32_k<<<smBlocks, 256, 0, stream>>>(out, N);
}